// ChamferPytorch_82575041233285
// MI455X (gfx1250) — compile-verified
//
#include <hip/hip_runtime.h>

typedef __attribute__((ext_vector_type(2))) float v2f;
typedef __attribute__((ext_vector_type(8))) float v8f;

#define CH 4  // chunks per B-sweep (parallelism multiplier)

// ---------------------------------------------------------------------------
// Per point set, build:
//  pk[i] = (p0, p1, p2, |p|^2)              (A-fragment source + q^2 epilogue)
//  bf    = pre-swizzled WMMA B fragments, pair-interleaved so one b128 load
//          per lane feeds TWO 16x16x4 WMMAs:
//    float4 slot (t*64 + lane):  .xy = frag of tile 2t, .zw = frag of tile 2t+1
//    lane l<16  (K=0,1): (-2*p0, -2*p1)
//    lane l>=16 (K=2,3): (-2*p2, |p|^2)
//  => V_WMMA_F32_16X16X4_F32(A,B,0)[m,n] = |p_n|^2 - 2*q_m.p_n   directly.
// ---------------------------------------------------------------------------
__global__ void chamfer_pack(const float* __restrict__ p,
                             float4* __restrict__ pk,
                             float2* __restrict__ bf,
                             int n) {
    int i = blockIdx.x * blockDim.x + threadIdx.x;
    if (i >= n) return;
    float a = p[3 * i + 0], b = p[3 * i + 1], c = p[3 * i + 2];
    float sq = a * a + b * b + c * c;
    pk[i] = make_float4(a, b, c, sq);
    int j = i >> 4, l = i & 15;
    int t = j >> 1, which = j & 1;                      // tile pair, slot half
    bf[(t * 64 + l) * 2 + which]      = make_float2(-2.0f * a, -2.0f * b);
    bf[(t * 64 + 16 + l) * 2 + which] = make_float2(-2.0f * c, sq);
}

// ---------------------------------------------------------------------------
// Fused both-direction pass. Wave-level parameters are forced into SGPRs via
// readfirstlane so the inner loop gets scalar control flow (EXEC stays all-1s
// for WMMA, and the unroller/clause-former can batch the loads).
// Inner loop per tile pair: 1 global_load_b128 + 2 WMMA + 8 v_min3.
// Writes 16 partial row-mins (pre-q^2, pre-clamp) per wave.
// ---------------------------------------------------------------------------
__global__ void __launch_bounds__(256)
chamfer_pass(const float4* __restrict__ xp,
             const float4* __restrict__ yp,
             const float2* __restrict__ xb,
             const float2* __restrict__ yb,
             int tilesX, int tilesY,
             float* __restrict__ pmin) {
    const int lane = threadIdx.x & 31;
    const int half = lane >> 4;
    const int l16  = lane & 15;

    // Scalar (SGPR) wave id: readfirstlane(threadIdx.x) is the wave's base tid.
    const int wib = (int)(__builtin_amdgcn_readfirstlane(threadIdx.x) >> 5);
    const int w   = (int)blockIdx.x * 8 + wib;          // scalar global wave id

    const float4* apack;
    const float2* bfmt;
    int nTilesB, qtile, chunk;
    const int dir0 = tilesX * CH;
    if (w < dir0) {               // x -> y  (scalar branch)
        apack = xp; bfmt = yb; nTilesB = tilesY;
        qtile = w / CH; chunk = w % CH;
    } else {                      // y -> x
        int w2 = w - dir0;
        apack = yp; bfmt = xb; nTilesB = tilesX;
        qtile = w2 / CH; chunk = w2 % CH;
    }

    // A fragment: K=0/1 in lower lanes, K=2/3 in upper; K=3 element = 1.0.
    float4 pa = apack[qtile * 16 + l16];
    v2f a;
    a.x = half ? pa.z : pa.x;
    a.y = half ? 1.0f : pa.y;

    v8f acc;
#pragma unroll
    for (int v = 0; v < 8; ++v) acc[v] = 3.4e38f;

    const int pairsPerChunk = (nTilesB >> 1) / CH;      // scalar trip count (128)
    const float4* bp = (const float4*)bfmt
                     + (size_t)(chunk * pairsPerChunk) * 64 + lane;

#pragma unroll 2
    for (int t = 0; t < pairsPerChunk; ++t) {
        float4 bv = bp[(size_t)t * 64];
        v2f b0; b0.x = bv.x; b0.y = bv.y;               // tile 2t fragment
        v2f b1; b1.x = bv.z; b1.y = bv.w;               // tile 2t+1 fragment
        v8f c = {};
        v8f d0 = __builtin_amdgcn_wmma_f32_16x16x4_f32(
            false, a, false, b0, (short)0, c, false, false);
        v8f d1 = __builtin_amdgcn_wmma_f32_16x16x4_f32(
            false, a, false, b1, (short)0, c, false, false);
#pragma unroll
        for (int v = 0; v < 8; ++v)
            acc[v] = fminf(fminf(acc[v], d0[v]), d1[v]);    // -> v_min3_num_f32
    }

    // Row-min: butterfly min across the 16 lanes of each half-wave.
    // Lower half ends with rows 0..7 in acc[0..7], upper half rows 8..15.
#pragma unroll
    for (int v = 0; v < 8; ++v) {
        float m = acc[v];
        m = fminf(m, __shfl_xor(m, 1, 32));
        m = fminf(m, __shfl_xor(m, 2, 32));
        m = fminf(m, __shfl_xor(m, 4, 32));
        m = fminf(m, __shfl_xor(m, 8, 32));
        acc[v] = m;
    }

    if (l16 == 0) {
        float4 r0 = make_float4(acc[0], acc[1], acc[2], acc[3]);
        float4 r1 = make_float4(acc[4], acc[5], acc[6], acc[7]);
        float4* dst = (float4*)(pmin + (size_t)w * 16 + half * 8);
        dst[0] = r0;
        dst[1] = r1;
    }
}

// ---------------------------------------------------------------------------
// Stage 2: min across the CH chunk partials per row, apply deferred |q|^2 and
// clamp:  rowval = max( |q|^2 + min_chunks(partial), 0 ).
// ---------------------------------------------------------------------------
__global__ void chamfer_rows(const float4* __restrict__ xp,
                             const float4* __restrict__ yp,
                             const float* __restrict__ pmin,
                             float* __restrict__ rowvals,
                             int N, int M, int tilesX) {
    int r = blockIdx.x * blockDim.x + threadIdx.x;
    if (r >= N + M) return;
    int dir = (r < N) ? 0 : 1;
    int m   = dir ? (r - N) : r;
    int qtile = m >> 4, row = m & 15;
    int base  = (dir ? tilesX * CH : 0) + qtile * CH;
    float v = 3.4e38f;
#pragma unroll
    for (int c = 0; c < CH; ++c)
        v = fminf(v, pmin[(size_t)(base + c) * 16 + row]);
    float q2 = dir ? yp[m].w : xp[m].w;
    rowvals[r] = fmaxf(v + q2, 0.0f);
}

// ---------------------------------------------------------------------------
// Deterministic final reduction: sum per-row NN distances, scale by 1/(N+M).
// ---------------------------------------------------------------------------
__global__ void chamfer_reduce(const float* __restrict__ rowvals, int n,
                               float* __restrict__ out, float invCount) {
    __shared__ float sm[256];
    float s = 0.0f;
    for (int i = threadIdx.x; i < n; i += 256) s += rowvals[i];
    sm[threadIdx.x] = s;
    __syncthreads();
    for (int off = 128; off > 0; off >>= 1) {
        if ((int)threadIdx.x < off) sm[threadIdx.x] += sm[threadIdx.x + off];
        __syncthreads();
    }
    if (threadIdx.x == 0) out[0] = sm[0] * invCount;
}

extern "C" void kernel_launch(void* const* d_in, const int* in_sizes, int n_in,
                              void* d_out, int out_size, void* d_ws, size_t ws_size,
                              hipStream_t stream) {
    (void)n_in; (void)out_size; (void)ws_size;
    const float* x = (const float*)d_in[0];
    const float* y = (const float*)d_in[1];
    const int N = in_sizes[0] / 3;   // 16384
    const int M = in_sizes[1] / 3;   // 16384

    const int tilesX = N / 16;       // 1024
    const int tilesY = M / 16;       // 1024

    // Workspace layout (all region starts 16B-aligned):
    float4* xp = (float4*)d_ws;                          //  N float4   (256 KB)
    float4* yp = xp + N;                                 //  M float4   (256 KB)
    float2* xb = (float2*)(yp + M);                      // 2N float2   (256 KB)
    float2* yb = xb + 2 * N;                             // 2M float2   (256 KB)
    float*  pmin = (float*)(yb + 2 * M);                 // (tX+tY)*CH*16 (512 KB)
    float*  rowvals = pmin + (size_t)(tilesX + tilesY) * CH * 16;  // N+M (128 KB)

    chamfer_pack<<<(N + 255) / 256, 256, 0, stream>>>(x, xp, xb, N);
    chamfer_pack<<<(M + 255) / 256, 256, 0, stream>>>(y, yp, yb, M);

    // Fused both directions, CH chunks each: 8192 waves in one launch.
    const int totalWaves = (tilesX + tilesY) * CH;
    chamfer_pass<<<totalWaves / 8, 256, 0, stream>>>(xp, yp, xb, yb,
                                                     tilesX, tilesY, pmin);

    chamfer_rows<<<(N + M + 255) / 256, 256, 0, stream>>>(xp, yp, pmin, rowvals,
                                                          N, M, tilesX);

    chamfer_reduce<<<1, 256, 0, stream>>>(rowvals, N + M, (float*)d_out,
                                          1.0f / (float)(N + M));
}